// PosEncoding_41042707481087
// MI455X (gfx1250) — compile-verified
//
#include <hip/hip_runtime.h>
#include <stdint.h>

// Positional encoding: out[row] = [x0,x1,x2, {sin(2^i*pi*xj), cos(2^i*pi*xj)}_{i<10,j<3}]
// Memory-bound: ~528 MB traffic -> ~23us floor at 23.3 TB/s.
//
// v_sin_f32/v_cos_f32 take a revolutions argument (radians * 1/2pi), so
// sin(2^i*pi*x) == v_sin(2^(i-1)*x): start r = 0.5*x and double each step.
//
// Output rows are 63 floats (odd), so we stage each block's 256x63 tile in LDS
// (stride 63 vs 64 banks -> conflict free) and stream it out with the CDNA5
// async LDS->global DMA path (global_store_async_from_lds_b128 + ASYNCcnt),
// marked non-temporal: the 504 MB write stream is never re-read and should
// not displace L2 contents.

#define IN_F   3
#define NF     10
#define OUT_D  63
#define BLOCK  256
#define ROWS   256   // rows per block == one thread per row

__global__ __launch_bounds__(BLOCK) void posenc_kernel(const float* __restrict__ coords,
                                                       float* __restrict__ out,
                                                       int n) {
  __shared__ float s_out[ROWS * OUT_D];   // 64512 bytes, 16B-aligned tile

  const int tid = threadIdx.x;
  const long long row0 = (long long)blockIdx.x * ROWS;
  const long long row  = row0 + tid;

  // ---- compute: one row per thread, results into LDS ----
  if (row < (long long)n) {
    const float x0 = coords[row * IN_F + 0];
    const float x1 = coords[row * IN_F + 1];
    const float x2 = coords[row * IN_F + 2];
    float* o = &s_out[tid * OUT_D];
    o[0] = x0; o[1] = x1; o[2] = x2;
    // revolutions argument: r_i = 2^(i-1) * x  (exact power-of-two scaling)
    float r0 = x0 * 0.5f, r1 = x1 * 0.5f, r2 = x2 * 0.5f;
#pragma unroll
    for (int i = 0; i < NF; ++i) {
      float* p = o + 3 + i * 6;
      p[0] = __builtin_amdgcn_sinf(r0);   // v_sin_f32 (revolutions in)
      p[1] = __builtin_amdgcn_cosf(r0);   // v_cos_f32
      p[2] = __builtin_amdgcn_sinf(r1);
      p[3] = __builtin_amdgcn_cosf(r1);
      p[4] = __builtin_amdgcn_sinf(r2);
      p[5] = __builtin_amdgcn_cosf(r2);
      r0 += r0; r1 += r1; r2 += r2;       // double the angle each frequency
    }
  }
  __syncthreads();

  // ---- stream tile out: async LDS -> global, 16B per lane per issue ----
  const long long rem = (long long)n - row0;
  const int rows_here = (int)(rem < (long long)ROWS ? rem : (long long)ROWS);
  const int nfloat = rows_here * OUT_D;
  const int nvec   = nfloat >> 2;                    // float4 chunks
  float* gbase = out + row0 * OUT_D;                 // 16B aligned (256*63*4 % 16 == 0)
  // low 32 bits of a generic LDS pointer are the LDS byte address
  const unsigned lds_base = (unsigned)(uintptr_t)&s_out[0];

  for (int v = tid; v < nvec; v += BLOCK) {
    unsigned long long gaddr = (unsigned long long)(uintptr_t)(gbase + (size_t)v * 4);
    unsigned laddr = lds_base + (unsigned)v * 16u;
    asm volatile("global_store_async_from_lds_b128 %0, %1, off th:TH_STORE_NT"
                 :
                 : "v"(gaddr), "v"(laddr)
                 : "memory");
  }
  // scalar cleanup for any non-multiple-of-4 tail (none for this N, but be safe)
  for (int k = (nvec << 2) + tid; k < nfloat; k += BLOCK) {
    gbase[k] = s_out[k];
  }
  asm volatile("s_wait_asynccnt 0x0" ::: "memory");
}

extern "C" void kernel_launch(void* const* d_in, const int* in_sizes, int n_in,
                              void* d_out, int out_size, void* d_ws, size_t ws_size,
                              hipStream_t stream) {
  (void)n_in; (void)out_size; (void)d_ws; (void)ws_size;
  const float* coords = (const float*)d_in[0];
  float* out = (float*)d_out;
  const int n = in_sizes[0] / IN_F;                 // number of rows (2,000,000)
  const int grid = (n + ROWS - 1) / ROWS;           // 7813 blocks
  posenc_kernel<<<grid, BLOCK, 0, stream>>>(coords, out, n);
}